// LearnableTensorSquare_2911987826946
// MI455X (gfx1250) — compile-verified
//
#include <hip/hip_runtime.h>
#include <hip/hip_bf16.h>

// Problem constants (match the reference).
#define Zn 131072
#define Nn 64
#define Kn 64
#define Wn 32

typedef __attribute__((ext_vector_type(2))) float v2f;
typedef __attribute__((ext_vector_type(8))) float v8f;

// ---------------------------------------------------------------------------
// Kernel 1: M[k,i,j] = sum_w weight[w] * mix[w,k,i,j], written in the exact
// per-lane swizzled layout the WMMA B-fragment loads want:
//   B-frag for (ktile, i, jj): lane L holds float2
//     { M[k, i, j0 + 2*half], M[k, i, j0 + 2*half + 1] }
//   with k = ktile*16 + (L&15), j0 = jj*4, half = L>>4
// (32-bit B 4x16 layout: VGPR0 = K rows 0/2 across lane halves, VGPR1 = 1/3).
// Reads of `mix` are fully coalesced (consecutive t -> consecutive addrs).
// ---------------------------------------------------------------------------
__global__ __launch_bounds__(256) void lts_reduce_mix_kernel(
    const float* __restrict__ mix,   // [W, K*N*N]
    const float* __restrict__ wgt,   // [W]
    float* __restrict__ Msw)         // [K*N*N] swizzled, in d_ws
{
    const int t = blockIdx.x * blockDim.x + threadIdx.x;   // 0 .. 262143
    float acc = 0.0f;
#pragma unroll
    for (int w = 0; w < Wn; ++w)
        acc = fmaf(wgt[w], mix[w * (Kn * Nn * Nn) + t], acc);

    const int k  = t >> 12;          // [0,64)
    const int i  = (t >> 6) & 63;    // [0,64)
    const int j  = t & 63;           // [0,64)
    const int kt = k >> 4;           // ktile
    const int kl = k & 15;
    const int jj = j >> 2;           // reduction step within i
    const int jo = j & 3;
    const int half = jo >> 1;
    const int lane = half * 16 + kl;
    // float2 element index, then select .x/.y via (jo&1)
    const int fidx = (((kt * 64 + i) * 16 + jj) * 32 + lane) * 2 + (jo & 1);
    Msw[fidx] = acc;
}

// ---------------------------------------------------------------------------
// Kernel 2: out[z,k] = sum_{i,j} M[k,i,j] f[z,i] f[z,j]
// Block = 1024 threads = 32 waves; each wave owns 16 z-rows (block: 512 rows).
// Outer loop over 4 k-tiles: block stages the 256 KB M-slice for that k-tile
// into LDS once (shared by all 32 waves -> M global traffic = 256 blocks * 1MB
// = 256 MB, all L2-resident), then each wave runs the i/jj reduction:
//   per step: ds_load_b64 (B frag) + v_mul (build G frag) + wmma f32 16x16x4.
// A-side feature values live in registers (16 x float2 per lane, loaded once).
// ---------------------------------------------------------------------------
__global__ __launch_bounds__(1024) void lts_bilinear_wmma_kernel(
    const float* __restrict__ feat,  // [Z, 64]
    const float* __restrict__ Msw,   // swizzled M from kernel 1
    float* __restrict__ out)         // [Z, 64]
{
    extern __shared__ float smem[];                // 256 KB: one k-tile of M

    const int tid  = threadIdx.x;
    const int wave = tid >> 5;
    const int lane = tid & 31;
    const int half = lane >> 4;       // 0: K rows 0/1, 1: K rows 2/3
    const int l16  = lane & 15;
    const int zrow = blockIdx.x * 512 + wave * 16 + l16;  // A-side row for lane

    // Load this lane's feature data for all 16 jj steps (j0 = jj*4 + 2*half).
    // 8-byte aligned (even element offsets). Reused across all i / k-tiles.
    v2f  fj[16];
#pragma unroll
    for (int jj = 0; jj < 16; ++jj) {
        const float2 v = *(const float2*)(feat + zrow * 64 + jj * 4 + half * 2);
        fj[jj].x = v.x;
        fj[jj].y = v.y;
    }
    // Full feature row for the f[z,i] factor (read via cache, per-lane scalar).
    const float* frow = feat + zrow * 64;

    for (int kt = 0; kt < 4; ++kt) {
        __syncthreads();   // previous k-tile's reads done before overwrite
        // Stage 256 KB M-slice: 16384 float4, 1024 threads -> 16 iters, coalesced.
        {
            const float4* src = (const float4*)(Msw + (size_t)kt * 65536);
            float4* dst = (float4*)smem;
            for (int idx = tid; idx < 16384; idx += 1024)
                dst[idx] = src[idx];
        }
        __syncthreads();

        const float2* Bl = (const float2*)smem;
        v8f acc = {};
        for (int i = 0; i < 64; ++i) {
            const float fi = frow[i];                       // f[z_lane, i]
            const float2* bp = Bl + i * 512 + lane;
#pragma unroll
            for (int jj = 0; jj < 16; ++jj) {
                // B fragment straight from LDS (pre-swizzled layout).
                const float2 bv = bp[jj * 32];
                v2f b; b.x = bv.x; b.y = bv.y;
                // A fragment of G = f[z,i]*f[z,j..]: one packed mul.
                v2f a; a.x = fi * fj[jj].x; a.y = fi * fj[jj].y;
                acc = __builtin_amdgcn_wmma_f32_16x16x4_f32(
                    /*neg_a=*/false, a, /*neg_b=*/false, b,
                    /*c_mod=*/(short)0, acc,
                    /*reuse_a=*/false, /*reuse_b=*/false);
            }
        }

        // D layout: VGPR r -> row r + 8*half (within wave's 16-row tile),
        // col = lane&15 within this k-tile.
#pragma unroll
        for (int r = 0; r < 8; ++r) {
            const int z = blockIdx.x * 512 + wave * 16 + half * 8 + r;
            out[z * 64 + kt * 16 + l16] = acc[r];
        }
    }
}

// ---------------------------------------------------------------------------
extern "C" void kernel_launch(void* const* d_in, const int* in_sizes, int n_in,
                              void* d_out, int out_size, void* d_ws, size_t ws_size,
                              hipStream_t stream) {
    const float* feat = (const float*)d_in[0];   // [Z, N] f32
    const float* mix  = (const float*)d_in[1];   // [W, K, N, N] f32
    const float* wgt  = (const float*)d_in[2];   // [W] f32
    float* out = (float*)d_out;                  // [Z, K] f32
    float* Msw = (float*)d_ws;                   // 1 MB swizzled M

    // Kernel 1: 262144 elements, 256 threads/block.
    lts_reduce_mix_kernel<<<(Kn * Nn * Nn) / 256, 256, 0, stream>>>(mix, wgt, Msw);

    // Kernel 2: 512 z-rows per block, 256 KB dynamic LDS for the M k-tile.
    lts_bilinear_wmma_kernel<<<Zn / 512, 1024, 65536 * sizeof(float), stream>>>(
        feat, Msw, out);
}